// GAT_51977694216499
// MI455X (gfx1250) — compile-verified
//
#include <hip/hip_runtime.h>
#include <hip/hip_bf16.h>
#include <math.h>

typedef __attribute__((ext_vector_type(2))) float v2f;
typedef __attribute__((ext_vector_type(8))) float v8f;

#define NUM_GRAPHS 64
#define NEG_SLOPE 0.2f

// ---------------------------------------------------------------------------
// Dense projection h = x @ W via V_WMMA_F32_16X16X4_F32.
// One wave -> 16 rows x FOUT cols. FOUT/16 accumulators, K stepped by 4.
// A layout (16x4 f32): lanes 0-15 = rows, VGPR0={K0|K2}, VGPR1={K1|K3}.
// B layout (4x16 f32): lanes 0-15 = cols, VGPR0={K0|K2}, VGPR1={K1|K3}.
// C/D (16x16 f32): VGPR i -> row i (lanes 0-15) / row 8+i (lanes 16-31).
// ---------------------------------------------------------------------------
template <int FIN, int FOUT>
__global__ __launch_bounds__(256) void gemm_wmma(const float* __restrict__ x,
                                                 const float* __restrict__ W,
                                                 float* __restrict__ h, int N) {
  constexpr int NT = FOUT / 16;
  __shared__ float Ws[FIN * FOUT];
  for (int i = threadIdx.x; i < FIN * FOUT; i += 256) Ws[i] = W[i];
  __syncthreads();

  const int wave = threadIdx.x >> 5;
  const int lane = threadIdx.x & 31;
  const int m0 = (blockIdx.x * 8 + wave) * 16;
  const int halfsel = lane >> 4;  // 0 or 1
  const int lidx = lane & 15;
  int row = m0 + lidx;
  if (row >= N) row = N - 1;      // lane-select only; EXEC stays all-1s
  const int khalf = halfsel * 2;

  v8f acc[NT];
#pragma unroll
  for (int t = 0; t < NT; ++t)
#pragma unroll
    for (int j = 0; j < 8; ++j) acc[t][j] = 0.0f;

#pragma unroll 4
  for (int k = 0; k < FIN; k += 4) {
    const int kk = k + khalf;
    v2f a;
    a.x = x[(size_t)row * FIN + kk];
    a.y = x[(size_t)row * FIN + kk + 1];
#pragma unroll
    for (int t = 0; t < NT; ++t) {
      v2f b;
      b.x = Ws[kk * FOUT + t * 16 + lidx];
      b.y = Ws[(kk + 1) * FOUT + t * 16 + lidx];
      acc[t] = __builtin_amdgcn_wmma_f32_16x16x4_f32(
          false, a, false, b, (short)0, acc[t], false, false);
    }
  }

  const int rbase = m0 + halfsel * 8;
#pragma unroll
  for (int i = 0; i < 8; ++i) {
    const int r = rbase + i;
    if (r < N) {
#pragma unroll
      for (int t = 0; t < NT; ++t)
        h[(size_t)r * FOUT + t * 16 + lidx] = acc[t][i];
    }
  }
}

// s[n] = h[n,:] . a_src ; d[n] = h[n,:] . a_dst
template <int F>
__global__ void node_dots(const float* __restrict__ h,
                          const float* __restrict__ asrc,
                          const float* __restrict__ adst,
                          float* __restrict__ s, float* __restrict__ d, int N) {
  int n = blockIdx.x * blockDim.x + threadIdx.x;
  if (n >= N) return;
  float ss = 0.0f, dd = 0.0f;
#pragma unroll
  for (int f = 0; f < F; ++f) {
    float v = h[(size_t)n * F + f];
    ss += v * asrc[f];
    dd += v * adst[f];
  }
  s[n] = ss;
  d[n] = dd;
}

__device__ __forceinline__ void atomicMaxFloat(float* addr, float val) {
  int vi = __float_as_int(val);
  if (vi >= 0)
    atomicMax((int*)addr, vi);
  else
    atomicMin((unsigned int*)addr, (unsigned int)vi);
}

// Pass 1: e = leaky_relu(s[src] + d[dst]); running max per dst.
__global__ void edge_score(const int* __restrict__ src,
                           const int* __restrict__ dst,
                           const float* __restrict__ sv,
                           const float* __restrict__ dv,
                           float* __restrict__ ebuf, float* __restrict__ m,
                           int E, int ET) {
  int id = blockIdx.x * blockDim.x + threadIdx.x;
  if (id >= ET) return;
  int si, di;
  if (id < E) {
    si = src[id];
    di = dst[id];
  } else {
    si = di = id - E;  // self-loop
  }
  float e = sv[si] + dv[di];
  e = (e > 0.0f) ? e : NEG_SLOPE * e;
  ebuf[id] = e;
  atomicMaxFloat(m + di, e);
}

// Pass 2: t = exp(e - m[dst]); z[dst] += t.
__global__ void edge_exp(const int* __restrict__ dst,
                         const float* __restrict__ m, float* __restrict__ ebuf,
                         float* __restrict__ z, int E, int ET) {
  int id = blockIdx.x * blockDim.x + threadIdx.x;
  if (id >= ET) return;
  int di = (id < E) ? dst[id] : id - E;
  float t = __expf(ebuf[id] - m[di]);
  ebuf[id] = t;
  atomicAdd(z + di, t);
}

// Pass 3: out[dst,:] += (t / z[dst]) * h[src,:]. One wave per edge.
template <int F>
__global__ __launch_bounds__(256) void edge_aggregate(
    const int* __restrict__ src, const int* __restrict__ dst,
    const float* __restrict__ ebuf, const float* __restrict__ z,
    const float* __restrict__ h, float* __restrict__ out, int E, int ET) {
  int id = blockIdx.x * 8 + (threadIdx.x >> 5);
  if (id >= ET) return;
  int lane = threadIdx.x & 31;
  int si, di;
  if (id < E) {
    si = src[id];
    di = dst[id];
  } else {
    si = di = id - E;
  }
  float alpha = ebuf[id] / z[di];
#pragma unroll
  for (int f = lane; f < F; f += 32)
    atomicAdd(out + (size_t)di * F + f, alpha * h[(size_t)si * F + f]);
}

template <int F, bool ELU>
__global__ void bias_act(float* __restrict__ h, const float* __restrict__ b,
                         int N) {
  int i = blockIdx.x * blockDim.x + threadIdx.x;
  if (i >= N * F) return;
  float v = h[i] + b[i % F];
  if (ELU) v = (v > 0.0f) ? v : (__expf(v) - 1.0f);
  h[i] = v;
}

__global__ void pool_sum(const float* __restrict__ h,
                         const int* __restrict__ batch, float* __restrict__ out,
                         float* __restrict__ cnt, int N) {
  int i = blockIdx.x * blockDim.x + threadIdx.x;
  int n = i >> 6;
  int f = i & 63;
  if (n >= N) return;
  int g = batch[n];
  atomicAdd(out + g * 64 + f, h[(size_t)n * 64 + f]);
  if (f == 0) atomicAdd(cnt + g, 1.0f);
}

__global__ void pool_div(float* __restrict__ out, const float* __restrict__ cnt) {
  int i = blockIdx.x * blockDim.x + threadIdx.x;
  if (i >= NUM_GRAPHS * 64) return;
  float c = cnt[i >> 6];
  out[i] = out[i] / fmaxf(c, 1.0f);
}

__global__ void fill_kernel(float* __restrict__ p, float v, int n) {
  int i = blockIdx.x * blockDim.x + threadIdx.x;
  if (i < n) p[i] = v;
}

static inline int cdiv(int a, int b) { return (a + b - 1) / b; }

extern "C" void kernel_launch(void* const* d_in, const int* in_sizes, int n_in,
                              void* d_out, int out_size, void* d_ws,
                              size_t ws_size, hipStream_t stream) {
  const float* x      = (const float*)d_in[0];
  const int*   ei     = (const int*)d_in[1];
  const int*   batch  = (const int*)d_in[2];
  const float* W1     = (const float*)d_in[3];
  const float* a_src1 = (const float*)d_in[4];
  const float* a_dst1 = (const float*)d_in[5];
  const float* b1     = (const float*)d_in[6];
  const float* W2     = (const float*)d_in[7];
  const float* a_src2 = (const float*)d_in[8];
  const float* a_dst2 = (const float*)d_in[9];
  const float* b2     = (const float*)d_in[10];

  const int N = in_sizes[2];       // batch has one entry per node
  const int E = in_sizes[1] / 2;   // edge_index is [2, E]
  const int ET = E + N;            // + self-loops
  const int* src = ei;
  const int* dst = ei + E;

  float* ws = (float*)d_ws;
  size_t off = 0;
  float* h1 = ws + off; off += (size_t)N * 32;  // x @ W1
  float* o1 = ws + off; off += (size_t)N * 32;  // layer-1 aggregate -> elu -> layer-2 input
  float* h2 = ws + off; off += (size_t)N * 64;  // o1 @ W2
  float* o2 = ws + off; off += (size_t)N * 64;  // layer-2 aggregate
  float* sv = ws + off; off += (size_t)N;
  float* dv = ws + off; off += (size_t)N;
  float* mb = ws + off; off += (size_t)N;
  float* zb = ws + off; off += (size_t)N;
  float* eb = ws + off; off += (size_t)ET;
  float* cnt = ws + off; off += NUM_GRAPHS;

  const float NEG_INF = -__builtin_inff();

  // ---------------- Layer 1 ----------------
  gemm_wmma<128, 32><<<cdiv(N, 128), 256, 0, stream>>>(x, W1, h1, N);
  node_dots<32><<<cdiv(N, 256), 256, 0, stream>>>(h1, a_src1, a_dst1, sv, dv, N);
  fill_kernel<<<cdiv(N, 256), 256, 0, stream>>>(mb, NEG_INF, N);
  fill_kernel<<<cdiv(N, 256), 256, 0, stream>>>(zb, 0.0f, N);
  fill_kernel<<<cdiv(N * 32, 256), 256, 0, stream>>>(o1, 0.0f, N * 32);
  edge_score<<<cdiv(ET, 256), 256, 0, stream>>>(src, dst, sv, dv, eb, mb, E, ET);
  edge_exp<<<cdiv(ET, 256), 256, 0, stream>>>(dst, mb, eb, zb, E, ET);
  edge_aggregate<32><<<cdiv(ET, 8), 256, 0, stream>>>(src, dst, eb, zb, h1, o1, E, ET);
  bias_act<32, true><<<cdiv(N * 32, 256), 256, 0, stream>>>(o1, b1, N);

  // ---------------- Layer 2 ----------------
  gemm_wmma<32, 64><<<cdiv(N, 128), 256, 0, stream>>>(o1, W2, h2, N);
  node_dots<64><<<cdiv(N, 256), 256, 0, stream>>>(h2, a_src2, a_dst2, sv, dv, N);
  fill_kernel<<<cdiv(N, 256), 256, 0, stream>>>(mb, NEG_INF, N);
  fill_kernel<<<cdiv(N, 256), 256, 0, stream>>>(zb, 0.0f, N);
  fill_kernel<<<cdiv(N * 64, 256), 256, 0, stream>>>(o2, 0.0f, N * 64);
  edge_score<<<cdiv(ET, 256), 256, 0, stream>>>(src, dst, sv, dv, eb, mb, E, ET);
  edge_exp<<<cdiv(ET, 256), 256, 0, stream>>>(dst, mb, eb, zb, E, ET);
  edge_aggregate<64><<<cdiv(ET, 8), 256, 0, stream>>>(src, dst, eb, zb, h2, o2, E, ET);
  bias_act<64, false><<<cdiv(N * 64, 256), 256, 0, stream>>>(o2, b2, N);

  // ---------------- Global mean pool ----------------
  fill_kernel<<<cdiv(NUM_GRAPHS * 64, 256), 256, 0, stream>>>((float*)d_out, 0.0f,
                                                              NUM_GRAPHS * 64);
  fill_kernel<<<1, 64, 0, stream>>>(cnt, 0.0f, NUM_GRAPHS);
  pool_sum<<<cdiv(N * 64, 256), 256, 0, stream>>>(o2, batch, (float*)d_out, cnt, N);
  pool_div<<<cdiv(NUM_GRAPHS * 64, 256), 256, 0, stream>>>((float*)d_out, cnt);
}